// channel_cluster_layer_79757542686981
// MI455X (gfx1250) — compile-verified
//
#include <hip/hip_runtime.h>

typedef __attribute__((ext_vector_type(2))) float v2f;
typedef __attribute__((ext_vector_type(8))) float v8f;

#define B_   32
#define C_   2048
#define P_   8
#define L_   891
#define HID_ 8192

__device__ __forceinline__ float sigmoidf_(float v) {
    return 1.0f / (1.0f + __expf(-v));
}

// ---------------- Kernel 1: pooled[b][c] = mean over L of x[b][c][:] ----------------
// One wave (32 lanes) per (b,c) row; 8 waves per 256-thread block.
__global__ void pool_kernel(const float* __restrict__ x, float* __restrict__ pooled) {
    const int row  = blockIdx.x * 8 + (threadIdx.x >> 5);   // row in [0, B*C)
    const int lane = threadIdx.x & 31;
    const float* p = x + (size_t)row * L_;
    float s = 0.f;
    for (int i = lane; i < L_; i += 32) s += p[i];
    #pragma unroll
    for (int off = 16; off > 0; off >>= 1) s += __shfl_xor(s, off, 32);
    if (lane == 0) pooled[row] = s * (1.0f / (float)L_);
}

// ---------------- Kernels 2 & 3: out[m][n] = sigmoid(A[m,:] . W[n,:] + bias[n]) ------
// M fixed at 32 (2 row-tiles). One wave per 16x16 output tile.
// WMMA f32 16x16x4: lane holds (row = lane&15, hi = lane>>4), K index = 2*hi + vgpr,
// so both A and B fragments are one float2 load per lane per K-step of 4.
__global__ void gemm_bias_sigmoid(const float* __restrict__ A,
                                  const float* __restrict__ W,
                                  const float* __restrict__ bias,
                                  float* __restrict__ out,
                                  int N, int K) {
    const int wid    = blockIdx.x * (blockDim.x >> 5) + (threadIdx.x >> 5);
    const int lane   = threadIdx.x & 31;
    const int tile_m = wid & 1;        // 0..1  (M = 32)
    const int tile_n = wid >> 1;
    if (tile_n * 16 >= N) return;

    const int row = lane & 15;
    const int hi  = lane >> 4;

    const float* aptr = A + (size_t)(tile_m * 16 + row) * (size_t)K + 2 * hi;
    const float* bptr = W + (size_t)(tile_n * 16 + row) * (size_t)K + 2 * hi;

    v8f c0 = {};
    v8f c1 = {};
    // K is a multiple of 8 (2048 / 8192). Two accumulators hide WMMA->WMMA RAW.
    #pragma unroll 4
    for (int k0 = 0; k0 < K; k0 += 8) {
        v2f a0 = *(const v2f*)(aptr + k0);
        v2f b0 = *(const v2f*)(bptr + k0);
        v2f a1 = *(const v2f*)(aptr + k0 + 4);
        v2f b1 = *(const v2f*)(bptr + k0 + 4);
        c0 = __builtin_amdgcn_wmma_f32_16x16x4_f32(false, a0, false, b0,
                                                   (short)0, c0, false, false);
        c1 = __builtin_amdgcn_wmma_f32_16x16x4_f32(false, a1, false, b1,
                                                   (short)0, c1, false, false);
    }

    // C/D layout: VGPR v -> M = v + 8*hi, N = lane&15
    const int n    = tile_n * 16 + row;
    const float bv = bias[n];
    float* op = out + (size_t)(tile_m * 16 + 8 * hi) * (size_t)N + n;
    #pragma unroll
    for (int v = 0; v < 8; ++v) {
        float val = c0[v] + c1[v] + bv;
        op[(size_t)v * N] = sigmoidf_(val);
    }
}

// ---------------- Kernel 4: weighted[b,p,l] = (1/C) * sum_c g[b,p,c] * x[b,c,l] ------
// Block = 128 threads handles one (b, l-chunk of 128). Grouping weights staged in LDS
// transposed to [c][p] so the hot loop is 2 broadcast ds_load_b128 + 1 coalesced load.
__global__ void group_weight_kernel(const float* __restrict__ x,
                                    const float* __restrict__ g,
                                    float* __restrict__ out2) {
    __shared__ float wsm[C_ * P_];   // 64 KB: wsm[c*8 + p]
    const int b   = blockIdx.y;
    const int tid = threadIdx.x;     // 0..127

    const float* gb = g + (size_t)b * (P_ * C_);
    for (int idx = tid; idx < P_ * C_; idx += 128) {
        int p = idx >> 11;           // idx / 2048
        int c = idx & (C_ - 1);      // idx % 2048
        wsm[c * P_ + p] = gb[idx];
    }
    __syncthreads();

    const int  l     = blockIdx.x * 128 + tid;
    const bool valid = (l < L_);
    const float* xb  = x + (size_t)b * C_ * L_;

    float acc[P_] = {0.f, 0.f, 0.f, 0.f, 0.f, 0.f, 0.f, 0.f};
    for (int c = 0; c < C_; ++c) {
        float xv = valid ? xb[(size_t)c * L_ + l] : 0.f;
        const float4 w0 = *(const float4*)&wsm[c * P_];
        const float4 w1 = *(const float4*)&wsm[c * P_ + 4];
        acc[0] = fmaf(xv, w0.x, acc[0]);
        acc[1] = fmaf(xv, w0.y, acc[1]);
        acc[2] = fmaf(xv, w0.z, acc[2]);
        acc[3] = fmaf(xv, w0.w, acc[3]);
        acc[4] = fmaf(xv, w1.x, acc[4]);
        acc[5] = fmaf(xv, w1.y, acc[5]);
        acc[6] = fmaf(xv, w1.z, acc[6]);
        acc[7] = fmaf(xv, w1.w, acc[7]);
    }
    if (valid) {
        #pragma unroll
        for (int p = 0; p < P_; ++p)
            out2[((size_t)b * P_ + p) * L_ + l] = acc[p] * (1.0f / (float)C_);
    }
}

extern "C" void kernel_launch(void* const* d_in, const int* in_sizes, int n_in,
                              void* d_out, int out_size, void* d_ws, size_t ws_size,
                              hipStream_t stream) {
    (void)in_sizes; (void)n_in; (void)out_size; (void)ws_size;

    const float* x     = (const float*)d_in[0];   // (B, C, D, H, W)
    const float* fc1_w = (const float*)d_in[1];   // (HID, C)
    const float* fc1_b = (const float*)d_in[2];   // (HID)
    const float* fc2_w = (const float*)d_in[3];   // (P*C, HID)
    const float* fc2_b = (const float*)d_in[4];   // (P*C)

    float* g_out  = (float*)d_out;                        // (B, P, C)   = 524288 floats
    float* w_out  = g_out + (size_t)B_ * P_ * C_;         // (B, P, L)   = 228096 floats
    float* pooled = (float*)d_ws;                         // (B, C)      = 65536 floats
    float* h      = pooled + (size_t)B_ * C_;             // (B, HID)    = 262144 floats

    // 1) global average pool: 65536 rows, 8 waves/block
    pool_kernel<<<(B_ * C_) / 8, 256, 0, stream>>>(x, pooled);

    // 2) fc1 + sigmoid: tiles = 2 * (8192/16) = 1024 waves -> 256 blocks of 4 waves
    gemm_bias_sigmoid<<<(2 * (HID_ / 16)) / 4, 128, 0, stream>>>(
        pooled, fc1_w, fc1_b, h, HID_, C_);

    // 3) fc2 + sigmoid: tiles = 2 * (16384/16) = 2048 waves -> 512 blocks of 4 waves
    gemm_bias_sigmoid<<<(2 * ((P_ * C_) / 16)) / 4, 128, 0, stream>>>(
        h, fc2_w, fc2_b, g_out, P_ * C_, HID_);

    // 4) fused weighted channel reduction: grid (ceil(891/128)=7, B)
    group_weight_kernel<<<dim3(7, B_), 128, 0, stream>>>(x, g_out, w_out);
}